// PCF_utk_5068061409535
// MI455X (gfx1250) — compile-verified
//
#include <hip/hip_runtime.h>
#include <hip/hip_bf16.h>

// PCF histogram for MI455X (gfx1250, wave32).
// dist^2 via V_WMMA_F32_16X16X4_F32 rank-4 trick, Gaussian bins via v_exp_f32.

typedef float v2f __attribute__((ext_vector_type(2)));
typedef float v8f __attribute__((ext_vector_type(8)));

#define NBBINS 50

// ---- compile-time constants (clang constant-folds __builtin_sqrt of literals)
__device__ __forceinline__ double rmax_d() {
  return 2.0 * __builtin_sqrt(1.0 / (2.0 * __builtin_sqrt(3.0) * 100.0));
}

__global__ void pcf_zero_kernel(float* __restrict__ hist) {
  if (threadIdx.x < NBBINS) hist[threadIdx.x] = 0.0f;
}

__global__ __launch_bounds__(256) void pcf_hist_kernel(
    const float* __restrict__ da, const float* __restrict__ db,
    const int* __restrict__ samep, float* __restrict__ hist,
    int N, int M) {
  const int lane   = threadIdx.x & 31;
  const int wave   = (int)((blockIdx.x * blockDim.x + threadIdx.x) >> 5);
  const int nwaves = (int)((gridDim.x * blockDim.x) >> 5);

  const int same = samep[0];
  const float* __restrict__ bsrc = same ? da : db;

  const int tJ = M >> 4;              // 16x16 tiles along j
  const int NT = (N >> 4) * tJ;

  const int half = lane >> 4;         // 0: lanes 0-15, 1: lanes 16-31
  const int lm   = lane & 15;

  const double RMAXd = rmax_d();
  const double RAd   = 0.1 * RMAXd;           // 1 * (5/50) * RMAX
  const double RBd   = 5.0 * RMAXd;           // 50 * (5/50) * RMAX
  const double STEPd = (RBd - RAd) / (double)(NBBINS - 1);
  // s = sqrt(0.5/sigma^2 * log2(e)) so GF*exp(-0.5 x^2/sig^2) = GF*exp2(-(s*x)^2)
  const float sA = (float)__builtin_sqrt(0.5 / (0.01 * 0.01) * 1.4426950408889634);

  float h[NBBINS];
#pragma unroll
  for (int k = 0; k < NBBINS; ++k) h[k] = 0.0f;

  for (int t = wave; t < NT; t += nwaves) {
    const int i0 = (t / tJ) << 4;
    const int j0 = (t % tJ) << 4;

    // A fragment (16x4 f32): row m = [ax, ay, |a|^2, 1]
    // layout: v0 = K0 (lanes 0-15) / K2 (lanes 16-31); v1 = K1 / K3
    const float ax = da[3 * (i0 + lm) + 0];
    const float ay = da[3 * (i0 + lm) + 1];
    v2f afrag;
    afrag.x = half ? (ax * ax + ay * ay) : ax;
    afrag.y = half ? 1.0f : ay;

    // B fragment (4x16 f32): col n = [-2bx, -2by, 1, |b|^2]^T
    const float bx = bsrc[3 * (j0 + lm) + 0];
    const float by = bsrc[3 * (j0 + lm) + 1];
    v2f bfrag;
    bfrag.x = half ? 1.0f : (-2.0f * bx);
    bfrag.y = half ? (bx * bx + by * by) : (-2.0f * by);

    v8f c = {};
    // D = A x B + 0  ->  16x16 tile of squared distances
    c = __builtin_amdgcn_wmma_f32_16x16x4_f32(
        /*neg_a=*/false, afrag, /*neg_b=*/false, bfrag,
        /*c_mod=*/(short)0, c, /*reuse_a=*/false, /*reuse_b=*/false);

    // scaled distances; C/D layout: VGPR v -> M = v + 8*half, N = lm
    // raw v_sqrt_f32 (~1 ulp) is plenty for a sigma=0.01 Gaussian; clamp
    // tiny negative dist^2 from FP cancellation so no NaN reaches exp2.
    float sd[8];
#pragma unroll
    for (int v = 0; v < 8; ++v) {
      const float d2 = fmaxf(c[v], 0.0f);
      float dd = __builtin_amdgcn_sqrtf(d2);
      const int gi = i0 + v + (half << 3);
      const int gj = j0 + lm;
      if (same && (gi == gj)) dd = 1.0e9f;  // diagonal mask -> exp underflows to 0
      sd[v] = sA * dd;
    }

    // accumulate all 50 bins for this tile's 8 per-lane distances
#pragma unroll
    for (int k = 0; k < NBBINS; ++k) {
      const float srk = sA * (float)(RAd + (double)k * STEPd);
      float acc = h[k];
#pragma unroll
      for (int v = 0; v < 8; ++v) {
        const float x = srk - sd[v];
        acc += __builtin_amdgcn_exp2f(-(x * x));  // neg folds into v_exp_f32 src mod
      }
      h[k] = acc;
    }
  }

  // wave32 butterfly reduction per bin, one global atomic per wave per bin
#pragma unroll
  for (int k = 0; k < NBBINS; ++k) {
    float s = h[k];
    for (int off = 16; off > 0; off >>= 1) s += __shfl_xor(s, off, 32);
    if (lane == 0) atomicAdd(&hist[k], s);
  }
}

__global__ void pcf_final_kernel(const float* __restrict__ hist,
                                 float* __restrict__ out, int N, int M) {
  const int k = threadIdx.x;
  if (k >= NBBINS) return;
  const double RMAXd = rmax_d();
  const double RAd   = 0.1 * RMAXd;
  const double STEPd = (5.0 * RMAXd - RAd) / (double)(NBBINS - 1);
  const float  PI    = 3.14159265358979323846f;
  const float  GF    = (float)(1.0 / (__builtin_sqrt(2.0 * 3.141592653589793) * 0.01));

  const float r = (float)(RAd + (double)k * STEPd);
  const float cov = 1.0f - 2.0f * r / PI * 2.0f + r * r / PI;
  const float factor = 1.0f / (2.0f * PI * r * cov);
  const float pcf1 = hist[k] * GF * factor / ((float)N * (float)M);

  out[2 * k + 0] = r / (float)RMAXd;
  out[2 * k + 1] = pcf1;
}

extern "C" void kernel_launch(void* const* d_in, const int* in_sizes, int n_in,
                              void* d_out, int out_size, void* d_ws, size_t ws_size,
                              hipStream_t stream) {
  const float* da = (const float*)d_in[0];   // disks_a: (N,3) f32
  const float* db = (const float*)d_in[1];   // disks_b: (M,3) f32
  const int*   sc = (const int*)d_in[2];     // same_category scalar
  float* hist = (float*)d_ws;                // 50-float scratch accumulator
  float* out  = (float*)d_out;               // (50,2) f32

  const int N = in_sizes[0] / 3;
  const int M = in_sizes[1] / 3;

  pcf_zero_kernel<<<1, 64, 0, stream>>>(hist);
  pcf_hist_kernel<<<512, 256, 0, stream>>>(da, db, sc, hist, N, M);
  pcf_final_kernel<<<1, 64, 0, stream>>>(hist, out, N, M);
}